// GraphSageEdgeReprLayer_77567109366526
// MI455X (gfx1250) — compile-verified
//
#include <hip/hip_runtime.h>
#include <hip/hip_bf16.h>
#include <math.h>

typedef __attribute__((ext_vector_type(2))) float v2f;
typedef __attribute__((ext_vector_type(8))) float v8f;
typedef __attribute__((ext_vector_type(4))) float fvec4;   // native clang vector (nontemporal-compatible)

#define NN 50000
#define EE 800000
#define DD 64
#define NT (NN/16)   // 3125 row tiles (exact)
#define ET (EE/16)   // 50000 edge tiles (exact)
#define EPS_BN 1e-5f
#define EPS_NORM 1e-12f

// ---- CDNA5 async global->LDS staging (ASYNCcnt path), with sync fallback ----
#if defined(__has_builtin)
#  if __has_builtin(__builtin_amdgcn_global_load_async_to_lds_b128)
#    define USE_ASYNC_LDS 1
#  endif
#endif
#ifndef USE_ASYNC_LDS
#  define USE_ASYNC_LDS 0
#endif

#if USE_ASYNC_LDS
typedef int v4i __attribute__((vector_size(16)));
typedef __attribute__((address_space(1))) v4i* gas_v4i_p;   // global (device) AS
typedef __attribute__((address_space(3))) v4i* las_v4i_p;   // LDS (shared) AS

__device__ __forceinline__ void async_copy16(const float* gsrc, float* ldst) {
    __builtin_amdgcn_global_load_async_to_lds_b128(
        (gas_v4i_p)(gsrc), (las_v4i_p)(ldst), 0, 0);
}
__device__ __forceinline__ void async_wait() {
    asm volatile("s_wait_asynccnt 0x0" ::: "memory");
}
#endif

__device__ __forceinline__ v8f wmma_f32(v2f a, v2f b, v8f c) {
    // D = A(16x4,f32) * B(4x16,f32) + C(16x16,f32)
    return __builtin_amdgcn_wmma_f32_16x16x4_f32(
        /*neg_a=*/false, a, /*neg_b=*/false, b,
        /*c_mod=*/(short)0, c, /*reuse_a=*/false, /*reuse_b=*/false);
}

// ---------------------------------------------------------------- init
__global__ void k_init(float* cbuf, float* stats) {
    int i = blockIdx.x * blockDim.x + threadIdx.x;
    if (i < NN * DD) cbuf[i] = 0.0f;
    if (i < 256)     stats[i] = 0.0f;
}

// ---------------------------------------------------------------- node linear: Ah = h@A_W + A_b ; Bh = h@B_W + B_b
// block = 256 (8 waves): wave>>2 selects matrix (A/B), wave&3 selects 16-col tile
__global__ void k_node_linear(const float* __restrict__ h,
                              const float* __restrict__ A_W, const float* __restrict__ A_b,
                              const float* __restrict__ B_W, const float* __restrict__ B_b,
                              float* __restrict__ Ah, float* __restrict__ Bh) {
    __shared__ float tile[16 * DD];
    const int tid  = threadIdx.x;
    const int lane = tid & 31, half = lane >> 4, l16 = lane & 15;
    const int wave = tid >> 5;
    const int m    = wave >> 2;
    const int cb   = (wave & 3) * 16;
    const float* W    = m ? B_W : A_W;
    const float* bias = m ? B_b : A_b;
    float* out        = m ? Bh  : Ah;

    v2f bf[16];
#pragma unroll
    for (int s = 0; s < 16; ++s) {
        int k = 4 * s + 2 * half;
        bf[s].x = W[k * DD + cb + l16];
        bf[s].y = W[(k + 1) * DD + cb + l16];
    }
    const float bv = bias[cb + l16];

    for (int t = blockIdx.x; t < NT; t += gridDim.x) {
#if USE_ASYNC_LDS
        async_copy16(h + (size_t)t * 16 * DD + tid * 4, tile + tid * 4);
        async_wait();
#else
        ((fvec4*)tile)[tid] = ((const fvec4*)(h + (size_t)t * 16 * DD))[tid];
#endif
        __syncthreads();
        v8f acc = {};
#pragma unroll
        for (int s = 0; s < 16; ++s) {
            int k = 4 * s + 2 * half;
            v2f a; a.x = tile[l16 * DD + k]; a.y = tile[l16 * DD + k + 1];
            acc = wmma_f32(a, bf[s], acc);
        }
        // rows (t*16 + 8*half + r), fixed 256B stride -> immediate offsets
        float* outp = out + (size_t)(t * 16 + 8 * half) * DD + cb + l16;
#pragma unroll
        for (int r = 0; r < 8; ++r)
            outp[r * DD] = acc[r] + bv;
        __syncthreads();
    }
}

// ---------------------------------------------------------------- edge kernel
// Ce = e@C_W + C_b ; e_ij = Ce + Bh[src] + Bh[dst] ; msg = relu(sigmoid(e_ij)*Ah[src])
// segment-max via uint atomicMax (msg >= 0); BN stats of relu(e_ij)
__global__ void k_edge(const float* __restrict__ e,
                       const int* __restrict__ src, const int* __restrict__ dst,
                       const float* __restrict__ C_W, const float* __restrict__ C_b,
                       const float* __restrict__ Ah, const float* __restrict__ Bh,
                       float* __restrict__ eij_ws, unsigned int* __restrict__ cbuf,
                       float* __restrict__ e_sum, float* __restrict__ e_sumsq) {
    __shared__ float tile[16 * DD];
    __shared__ int sidx[16], didx[16];
    const int tid  = threadIdx.x;        // blockDim = 128 (4 waves)
    const int lane = tid & 31, half = lane >> 4, l16 = lane & 15;
    const int wave = tid >> 5;
    const int cb   = wave * 16;
    const int col  = cb + l16;

    v2f bf[16];
#pragma unroll
    for (int s = 0; s < 16; ++s) {
        int k = 4 * s + 2 * half;
        bf[s].x = C_W[k * DD + col];
        bf[s].y = C_W[(k + 1) * DD + col];
    }
    const float bv = C_b[col];
    float es = 0.0f, esq = 0.0f;

    for (int t = blockIdx.x; t < ET; t += gridDim.x) {
#if USE_ASYNC_LDS
        async_copy16(e + (size_t)t * 16 * DD + tid * 4,         tile + tid * 4);
        async_copy16(e + (size_t)t * 16 * DD + (tid + 128) * 4, tile + (tid + 128) * 4);
#else
        const fvec4* g4 = (const fvec4*)(e + (size_t)t * 16 * DD);
        ((fvec4*)tile)[tid]       = __builtin_nontemporal_load(&g4[tid]);
        ((fvec4*)tile)[tid + 128] = __builtin_nontemporal_load(&g4[tid + 128]);
#endif
        if (tid < 16) { sidx[tid] = src[t * 16 + tid]; didx[tid] = dst[t * 16 + tid]; }
#if USE_ASYNC_LDS
        async_wait();
#endif
        __syncthreads();
        v8f acc = {};
#pragma unroll
        for (int s = 0; s < 16; ++s) {
            int k = 4 * s + 2 * half;
            v2f a; a.x = tile[l16 * DD + k]; a.y = tile[l16 * DD + k + 1];
            acc = wmma_f32(a, bf[s], acc);
        }
        float* eijp = eij_ws + (size_t)(t * 16 + 8 * half) * DD + col;
#pragma unroll
        for (int r = 0; r < 8; ++r) {
            int rl = r + 8 * half;
            int sn = sidx[rl], dn = didx[rl];
            float eij = acc[r] + bv + Bh[(size_t)sn * DD + col] + Bh[(size_t)dn * DD + col];
            __builtin_nontemporal_store(eij, &eijp[r * DD]);   // streamed, keep L2 for gathers
            float rel = fmaxf(eij, 0.0f);
            es += rel; esq += rel * rel;
            float sg  = 1.0f / (1.0f + __expf(-eij));
            float msg = fmaxf(sg * Ah[(size_t)sn * DD + col], 0.0f);
            atomicMax(&cbuf[(size_t)dn * DD + col], __float_as_uint(msg));
        }
        __syncthreads();
    }
    atomicAdd(&e_sum[col], es);
    atomicAdd(&e_sumsq[col], esq);
}

// ---------------------------------------------------------------- apply: bundle=[h|c]@W_apply+b ; L2 norm ; relu ; BN stats
__global__ void k_apply(const float* __restrict__ h, const float* __restrict__ cbuf,
                        const float* __restrict__ W_apply, const float* __restrict__ b_apply,
                        float* __restrict__ h_pre,
                        float* __restrict__ h_sum, float* __restrict__ h_sumsq) {
    __shared__ float xt[16 * 128];
    __shared__ float bt[16 * DD];
    __shared__ float rn[16];
    const int tid  = threadIdx.x;        // blockDim = 128 (4 waves)
    const int lane = tid & 31, half = lane >> 4, l16 = lane & 15;
    const int wave = tid >> 5;
    const int cb   = wave * 16;

    v2f bf[32];
#pragma unroll
    for (int s = 0; s < 32; ++s) {
        int k = 4 * s + 2 * half;
        bf[s].x = W_apply[k * DD + cb + l16];
        bf[s].y = W_apply[(k + 1) * DD + cb + l16];
    }
    const float bv   = b_apply[cb + l16];
    const int scol   = tid & 63;
    const int rbase  = (tid >> 6) * 8;
    float hs = 0.0f, hsq = 0.0f;

    for (int t = blockIdx.x; t < NT; t += gridDim.x) {
#pragma unroll
        for (int j = 0; j < 8; ++j) {
            int q = tid + j * 128;               // 0..1023
            int row = q >> 6, c0 = q & 63;
            xt[row * 128 + c0]      = h[(size_t)(t * 16 + row) * DD + c0];
            xt[row * 128 + 64 + c0] = cbuf[(size_t)(t * 16 + row) * DD + c0];
        }
        __syncthreads();
        v8f acc = {};
#pragma unroll
        for (int s = 0; s < 32; ++s) {
            int k = 4 * s + 2 * half;
            v2f a; a.x = xt[l16 * 128 + k]; a.y = xt[l16 * 128 + k + 1];
            acc = wmma_f32(a, bf[s], acc);
        }
#pragma unroll
        for (int r = 0; r < 8; ++r)
            bt[(r + 8 * half) * DD + cb + l16] = acc[r] + bv;
        __syncthreads();
        if (tid < 16) {
            float s2 = 0.0f;
#pragma unroll
            for (int c0 = 0; c0 < DD; ++c0) { float v = bt[tid * DD + c0]; s2 += v * v; }
            rn[tid] = 1.0f / fmaxf(sqrtf(s2), EPS_NORM);
        }
        __syncthreads();
        float* hp = h_pre + (size_t)(t * 16 + rbase) * DD + scol;
#pragma unroll
        for (int r = 0; r < 8; ++r) {
            int row = rbase + r;
            float v = fmaxf(bt[row * DD + scol] * rn[row], 0.0f);
            hp[r * DD] = v;
            hs += v; hsq += v * v;
        }
        __syncthreads();
    }
    atomicAdd(&h_sum[scol], hs);
    atomicAdd(&h_sumsq[scol], hsq);
}

// ---------------------------------------------------------------- BN finalize + residual
__global__ void k_final_h(const float* __restrict__ h_in, const float* __restrict__ h_pre,
                          const float* __restrict__ h_sum, const float* __restrict__ h_sumsq,
                          const float* __restrict__ gamma, const float* __restrict__ beta,
                          float* __restrict__ h_out) {
    int i = blockIdx.x * blockDim.x + threadIdx.x;
    if (i >= NN * DD) return;
    int j = i & 63;
    float mu  = h_sum[j] * (1.0f / NN);
    float var = h_sumsq[j] * (1.0f / NN) - mu * mu;
    float inv = rsqrtf(var + EPS_BN);
    h_out[i] = h_in[i] + (h_pre[i] - mu) * inv * gamma[j] + beta[j];
}

__global__ void k_final_e(const float* __restrict__ e_in, const float* __restrict__ eij,
                          const float* __restrict__ e_sum, const float* __restrict__ e_sumsq,
                          const float* __restrict__ gamma, const float* __restrict__ beta,
                          float* __restrict__ e_out) {
    size_t i = (size_t)blockIdx.x * blockDim.x + threadIdx.x;
    if (i >= (size_t)EE * DD) return;
    int j = (int)(i & 63);
    float mu  = e_sum[j] * (1.0f / EE);
    float var = e_sumsq[j] * (1.0f / EE) - mu * mu;
    float inv = rsqrtf(var + EPS_BN);
    float v = fmaxf(__builtin_nontemporal_load(&eij[i]), 0.0f);
    float r = __builtin_nontemporal_load(&e_in[i]) + (v - mu) * inv * gamma[j] + beta[j];
    __builtin_nontemporal_store(r, &e_out[i]);
}

// ---------------------------------------------------------------- launcher
extern "C" void kernel_launch(void* const* d_in, const int* in_sizes, int n_in,
                              void* d_out, int out_size, void* d_ws, size_t ws_size,
                              hipStream_t stream) {
    const float* h   = (const float*)d_in[0];
    const float* e   = (const float*)d_in[1];
    const int*  src  = (const int*)d_in[2];
    const int*  dst  = (const int*)d_in[3];
    const float* A_W = (const float*)d_in[4];
    const float* A_b = (const float*)d_in[5];
    const float* B_W = (const float*)d_in[6];
    const float* B_b = (const float*)d_in[7];
    const float* C_W = (const float*)d_in[8];
    const float* C_b = (const float*)d_in[9];
    const float* W_apply = (const float*)d_in[10];
    const float* b_apply = (const float*)d_in[11];
    const float* gamma_h = (const float*)d_in[12];
    const float* beta_h  = (const float*)d_in[13];
    const float* gamma_e = (const float*)d_in[14];
    const float* beta_e  = (const float*)d_in[15];

    float* ws    = (float*)d_ws;
    float* Ah    = ws;
    float* Bh    = Ah    + (size_t)NN * DD;
    float* cbuf  = Bh    + (size_t)NN * DD;   // segment-max (uint bit patterns of >=0 floats)
    float* h_pre = cbuf  + (size_t)NN * DD;
    float* eij   = h_pre + (size_t)NN * DD;
    float* stats = eij   + (size_t)EE * DD;   // 256 floats
    float* h_sum = stats, *h_sumsq = stats + 64, *e_sum = stats + 128, *e_sumsq = stats + 192;

    float* h_out = (float*)d_out;
    float* e_out = h_out + (size_t)NN * DD;

    k_init<<<(NN * DD + 255) / 256, 256, 0, stream>>>(cbuf, stats);
    k_node_linear<<<1024, 256, 0, stream>>>(h, A_W, A_b, B_W, B_b, Ah, Bh);
    k_edge<<<2048, 128, 0, stream>>>(e, src, dst, C_W, C_b, Ah, Bh,
                                     eij, (unsigned int*)cbuf, e_sum, e_sumsq);
    k_apply<<<1024, 128, 0, stream>>>(h, cbuf, W_apply, b_apply, h_pre, h_sum, h_sumsq);
    k_final_h<<<(NN * DD + 255) / 256, 256, 0, stream>>>(h, h_pre, h_sum, h_sumsq,
                                                         gamma_h, beta_h, h_out);
    k_final_e<<<(int)(((size_t)EE * DD + 255) / 256), 256, 0, stream>>>(e, eij, e_sum, e_sumsq,
                                                                        gamma_e, beta_e, e_out);
}